// FraudGTLayer_35493609734438
// MI455X (gfx1250) — compile-verified
//
#include <hip/hip_runtime.h>
#include <hip/hip_bf16.h>
#include <math.h>

#define N_NODES 200000
#define N_EDGES 800000
#define DIN 64
#define DE  32
#define DH  64
#define NH  4
#define HD  16

typedef float v2f __attribute__((ext_vector_type(2)));
typedef float v8f __attribute__((ext_vector_type(8)));

__device__ __forceinline__ v8f wstep(v2f a, v2f b, v8f c) {
    // D = A(16x4 f32) * B(4x16 f32) + C, full fp32 precision (matches reference)
    return __builtin_amdgcn_wmma_f32_16x16x4_f32(false, a, false, b, (short)0, c,
                                                 false, false);
}

// C(16x16) = A(16xK, per-lane base ptr, row stride lda) * B(KxN row-major, col n0).
// All fragments are preloaded into registers first so the loads issue as one
// clause and the K/4 v_wmma instructions run back-to-back (accumulator chain
// is the only dependency).
template <int K>
__device__ __forceinline__ v8f gemm_tile(const float* As, int lda,
                                         const float* Bs, int ldb,
                                         int n0, int lane) {
    const int half = lane >> 4;
    const int qn   = lane & 15;
    v2f a[K / 4], b[K / 4];
    #pragma unroll
    for (int s = 0; s < K / 4; ++s) {
        const int k0 = 4 * s + 2 * half;
        a[s].x = As[qn * lda + k0];
        a[s].y = As[qn * lda + k0 + 1];
        b[s].x = Bs[k0 * ldb + n0 + qn];
        b[s].y = Bs[(k0 + 1) * ldb + n0 + qn];
    }
    v8f c = {0.f, 0.f, 0.f, 0.f, 0.f, 0.f, 0.f, 0.f};
    #pragma unroll
    for (int s = 0; s < K / 4; ++s) c = wstep(a[s], b[s], c);
    return c;
}

__global__ void k_zero(float* __restrict__ p, int n) {
    int i = blockIdx.x * blockDim.x + threadIdx.x;
    if (i < n) p[i] = 0.f;
}

// ---- Kernel 1: per-type LayerNorm + Q/K/V projections (WMMA) ----------------
// 2 waves/block (node-type boundary 100000 % 32 == 0 -> block-uniform type).
__global__ void k_node_qkv(const float* __restrict__ x,
                           const float* __restrict__ Wq, const float* __restrict__ bq,
                           const float* __restrict__ Wk, const float* __restrict__ bk,
                           const float* __restrict__ Wv, const float* __restrict__ bv,
                           const float* __restrict__ ln_s, const float* __restrict__ ln_b,
                           float* __restrict__ qo, float* __restrict__ ko,
                           float* __restrict__ vo) {
    __shared__ float xs[2][16 * 65];
    __shared__ float wlds[64 * 64];
    const int lane = threadIdx.x & 31;
    const int wv   = threadIdx.x >> 5;
    const int tile = blockIdx.x * 2 + wv;
    const int row0 = tile * 16;
    const int t    = (blockIdx.x * 32 >= N_NODES / 2) ? 1 : 0;
    float* As = xs[wv];

    for (int i = 0; i < 32; ++i) {               // cooperative 16x64 tile load
        int idx = i * 32 + lane;
        int m = idx >> 6, cc = idx & 63;
        As[m * 65 + cc] = x[(size_t)(row0 + m) * DIN + cc];
    }
    __syncthreads();
    {   // LayerNorm: lane pair (l, l+16) owns row l&15
        const int m = lane & 15, c0 = (lane >> 4) * 32;
        float s = 0.f;
        for (int j = 0; j < 32; ++j) s += As[m * 65 + c0 + j];
        s += __shfl_xor(s, 16, 32);
        float mean = s * (1.0f / 64.0f);
        float v2 = 0.f;
        for (int j = 0; j < 32; ++j) { float d = As[m * 65 + c0 + j] - mean; v2 += d * d; }
        v2 += __shfl_xor(v2, 16, 32);
        float inv = rsqrtf(v2 * (1.0f / 64.0f) + 1e-5f);
        for (int j = 0; j < 32; ++j) {
            int cc = c0 + j;
            As[m * 65 + cc] = (As[m * 65 + cc] - mean) * inv * ln_s[t * DIN + cc]
                              + ln_b[t * DIN + cc];
        }
    }

    const int half = lane >> 4, qn = lane & 15;
    const float* Ws[3] = {Wq + t * DIN * DH, Wk + t * DIN * DH, Wv + t * DIN * DH};
    const float* bs[3] = {bq + t * DH, bk + t * DH, bv + t * DH};
    float*       os[3] = {qo, ko, vo};
    for (int mm = 0; mm < 3; ++mm) {
        __syncthreads();                          // LN done / wlds reuse safe
        for (int i = threadIdx.x; i < DIN * DH; i += 64)
            wlds[i] = Ws[mm][i];                  // stage weights once per block
        __syncthreads();
        #pragma unroll
        for (int nt = 0; nt < 4; ++nt) {
            const int n0 = nt * 16;
            v8f c = gemm_tile<64>(As, 65, wlds, DH, n0, lane);
            float bias = bs[mm][n0 + qn];
            #pragma unroll
            for (int i = 0; i < 8; ++i) {
                int m = i + 8 * half;
                os[mm][(size_t)(row0 + m) * DH + n0 + qn] = c[i] + bias;
            }
        }
    }
}

// ---- Kernel 2: fused edge pass --------------------------------------------
// 4 waves/block (edge-type boundary 400000 % 64 == 0 -> block-uniform type).
// edge LN -> e_att/e_gate GEMMs (WMMA, weights in LDS) -> per-head key
// transform (gathered-A WMMA, ew^T in LDS) -> scores -> exp -> atomic scatter.
__global__ void k_edge(const float* __restrict__ ea,
                       const float* __restrict__ We, const float* __restrict__ be,
                       const float* __restrict__ Wg, const float* __restrict__ bg,
                       const float* __restrict__ ew,
                       const float* __restrict__ ln_s, const float* __restrict__ ln_b,
                       const int* __restrict__ esrc, const int* __restrict__ edst,
                       const float* __restrict__ qbuf, const float* __restrict__ kbuf,
                       const float* __restrict__ vbuf,
                       float* __restrict__ acc, float* __restrict__ den) {
    __shared__ float ebs[4][16 * 33];
    __shared__ float welds[DE * DH];              // We[t], row-major 32x64
    __shared__ float wglds[DE * DH];              // Wg[t]
    __shared__ float ewlds[NH * 16 * 17];         // ew[t]^T per head, stride 17
    __shared__ int   ssrc[4][16], sdst[4][16];
    const int lane = threadIdx.x & 31;
    const int wv   = threadIdx.x >> 5;
    const int tile = blockIdx.x * 4 + wv;
    const int e0   = tile * 16;
    const int t    = (blockIdx.x * 64 >= N_EDGES / 2) ? 1 : 0;
    float* As = ebs[wv];

    for (int i = 0; i < 16; ++i) {               // cooperative 16x32 tile load
        int idx = i * 32 + lane;
        int m = idx >> 5, cc = idx & 31;
        As[m * 33 + cc] = ea[(size_t)(e0 + m) * DE + cc];
    }
    if (lane < 16) {
        ssrc[wv][lane] = esrc[e0 + lane];
        sdst[wv][lane] = edst[e0 + lane];
    }
    // stage weights (block-wide)
    for (int i = threadIdx.x; i < DE * DH; i += 128) {
        welds[i] = We[t * DE * DH + i];
        wglds[i] = Wg[t * DE * DH + i];
    }
    for (int idx = threadIdx.x; idx < NH * 16 * 16; idx += 128) {
        int h = idx >> 8, rem = idx & 255, i = rem >> 4, j = rem & 15;
        ewlds[(h * 16 + j) * 17 + i] = ew[t * NH * 256 + idx];   // transposed
    }
    __syncthreads();
    {   // edge LayerNorm over 32 dims
        const int m = lane & 15, c0 = (lane >> 4) * 16;
        float s = 0.f;
        for (int j = 0; j < 16; ++j) s += As[m * 33 + c0 + j];
        s += __shfl_xor(s, 16, 32);
        float mean = s * (1.0f / 32.0f);
        float v2 = 0.f;
        for (int j = 0; j < 16; ++j) { float d = As[m * 33 + c0 + j] - mean; v2 += d * d; }
        v2 += __shfl_xor(v2, 16, 32);
        float inv = rsqrtf(v2 * (1.0f / 32.0f) + 1e-5f);
        for (int j = 0; j < 16; ++j) {
            int cc = c0 + j;
            As[m * 33 + cc] = (As[m * 33 + cc] - mean) * inv * ln_s[t * DE + cc]
                              + ln_b[t * DE + cc];
        }
    }
    __syncthreads();

    const int half = lane >> 4, qn = lane & 15;
    const float* krow = kbuf + (size_t)ssrc[wv][qn] * DH;  // A-frag row: edge qn

    for (int h = 0; h < NH; ++h) {
        const int n0 = h * 16;
        v8f eatt = gemm_tile<32>(As, 33, welds, DH, n0, lane);
        v8f egat = gemm_tile<32>(As, 33, wglds, DH, n0, lane);
        const float batt = be[t * DH + n0 + qn];
        const float bgat = bg[t * DH + n0 + qn];

        // ek' = (gathered k[src] 16edges x 16j) x ew[t][h]^T   (K=16 -> 4 WMMA)
        v8f ekc = gemm_tile<16>(krow + n0, 0, ewlds + h * 16 * 17, 17, 0, lane);

        // sv = q[dst] * ek' + e_att ; reduce over head dim (lanes%16 butterfly)
        v8f red;
        #pragma unroll
        for (int i = 0; i < 8; ++i) {
            int m = i + 8 * half;
            float dq = qbuf[(size_t)sdst[wv][m] * DH + n0 + qn];
            red[i] = dq * ekc[i] + eatt[i] + batt;
        }
        #pragma unroll
        for (int off = 1; off < 16; off <<= 1) {
            #pragma unroll
            for (int i = 0; i < 8; ++i) red[i] += __shfl_xor(red[i], off, 32);
        }
        float es8[8];
        #pragma unroll
        for (int i = 0; i < 8; ++i) {
            float sc = red[i] * 0.25f;                   // / sqrt(16)
            sc = fminf(5.0f, fmaxf(-5.0f, sc));          // clip => exp is safe,
            es8[i] = __expf(sc);                         // no segment-max pass
        }
        #pragma unroll
        for (int i = 0; i < 8; ++i) {                    // one writer per (edge,h)
            if (qn == i)
                atomicAdd(&den[(size_t)sdst[wv][i + 8 * half] * NH + h], es8[i]);
        }
        #pragma unroll
        for (int i = 0; i < 8; ++i) {                    // gated message scatter
            int m = i + 8 * half;
            float vv = vbuf[(size_t)ssrc[wv][m] * DH + n0 + qn];
            float g  = 1.0f / (1.0f + __expf(-(egat[i] + bgat)));
            atomicAdd(&acc[(size_t)sdst[wv][m] * DH + n0 + qn], es8[i] * vv * g);
        }
    }
}

// ---- Kernel 3: normalize + output projection (WMMA) ------------------------
// 2 waves/block (block-uniform node type).
__global__ void k_out(const float* __restrict__ acc, const float* __restrict__ den,
                      const float* __restrict__ Wo, const float* __restrict__ bo,
                      float* __restrict__ out) {
    __shared__ float ts[2][16 * 65];
    __shared__ float wlds[DH * DIN];
    const int lane = threadIdx.x & 31;
    const int wv   = threadIdx.x >> 5;
    const int tile = blockIdx.x * 2 + wv;
    const int row0 = tile * 16;
    const int t    = (blockIdx.x * 32 >= N_NODES / 2) ? 1 : 0;
    float* As = ts[wv];

    for (int i = threadIdx.x; i < DH * DIN; i += 64)
        wlds[i] = Wo[t * DH * DIN + i];           // stage Wo[t] once per block
    for (int i = 0; i < 32; ++i) {
        int idx = i * 32 + lane;
        int m = idx >> 6, cc = idx & 63;
        float d = den[(size_t)(row0 + m) * NH + (cc >> 4)];
        float a = acc[(size_t)(row0 + m) * DH + cc];
        As[m * 65 + cc] = (d > 0.f) ? (a / d) : 0.f;
    }
    __syncthreads();

    const int half = lane >> 4, qn = lane & 15;
    #pragma unroll
    for (int nt = 0; nt < 4; ++nt) {
        const int n0 = nt * 16;
        v8f c = gemm_tile<64>(As, 65, wlds, DIN, n0, lane);
        float bias = bo[t * DIN + n0 + qn];
        #pragma unroll
        for (int i = 0; i < 8; ++i)
            out[(size_t)(row0 + i + 8 * half) * DIN + n0 + qn] = c[i] + bias;
    }
}

extern "C" void kernel_launch(void* const* d_in, const int* in_sizes, int n_in,
                              void* d_out, int out_size, void* d_ws, size_t ws_size,
                              hipStream_t stream) {
    const float* x    = (const float*)d_in[0];
    const float* ea   = (const float*)d_in[1];
    const float* Wq   = (const float*)d_in[2];
    const float* bq   = (const float*)d_in[3];
    const float* Wk   = (const float*)d_in[4];
    const float* bk   = (const float*)d_in[5];
    const float* Wv   = (const float*)d_in[6];
    const float* bv   = (const float*)d_in[7];
    const float* Wo   = (const float*)d_in[8];
    const float* bo   = (const float*)d_in[9];
    const float* We   = (const float*)d_in[10];
    const float* be   = (const float*)d_in[11];
    const float* Wg   = (const float*)d_in[12];
    const float* bg   = (const float*)d_in[13];
    const float* ew   = (const float*)d_in[14];
    const float* lnns = (const float*)d_in[15];
    const float* lnnb = (const float*)d_in[16];
    const float* lnes = (const float*)d_in[17];
    const float* lneb = (const float*)d_in[18];
    const int*   ei   = (const int*)d_in[19];
    const int* esrc = ei;
    const int* edst = ei + N_EDGES;

    float* ws  = (float*)d_ws;
    float* qb  = ws;                                // q:   N x 64  (L2-resident)
    float* kb  = qb  + (size_t)N_NODES * DH;        // k:   N x 64
    float* vb  = kb  + (size_t)N_NODES * DH;        // v:   N x 64
    float* acc = vb  + (size_t)N_NODES * DH;        // acc: N x 64
    float* den = acc + (size_t)N_NODES * DH;        // den: N x 4

    const int nz = N_NODES * DH + N_NODES * NH;
    k_zero<<<(nz + 255) / 256, 256, 0, stream>>>(acc, nz);

    k_node_qkv<<<(N_NODES / 16) / 2, 64, 0, stream>>>(
        x, Wq, bq, Wk, bk, Wv, bv, lnns, lnnb, qb, kb, vb);

    k_edge<<<(N_EDGES / 16) / 4, 128, 0, stream>>>(
        ea, We, be, Wg, bg, ew, lnes, lneb, esrc, edst, qb, kb, vb, acc, den);

    k_out<<<(N_NODES / 16) / 2, 64, 0, stream>>>(acc, den, Wo, bo, (float*)d_out);
}